// SpatialRegionFocalLoss_39676907888505
// MI455X (gfx1250) — compile-verified
//
#include <hip/hip_runtime.h>

// Problem constants (shape (8, 256, 96, 96), REGION=3, BETA=2, EPS=1e-3)
#define BB   8
#define CC   256
#define HH   96
#define WW   96
#define HWSZ (HH * WW)       // 9216
#define NPIX (BB * HWSZ)     // 73728
#define NBW  32
#define NBLK 1024            // 32*32 blocks per batch
#define RR   3
#define EPSV 0.001f
#define BETA 2.0f

typedef float v2f __attribute__((ext_vector_type(2)));
typedef float v8f __attribute__((ext_vector_type(8)));

// ---------------------------------------------------------------------------
// Kernel 1: channel MSE reduction via V_WMMA_F32_16X16X4_F32.
// Each wave handles 16 consecutive pixels (same batch, same row: 96 % 16 == 0).
// A[m,k] = (rec-align)^2 for pixel m, channel c0+k;  B = ones(4x16);
// D = A*B + C accumulates the per-pixel channel sum in every column of D.
// A layout (32-bit 16x4): v0: lanes0-15 K=0, lanes16-31 K=2; v1: K=1 / K=3.
// D layout (32-bit 16x16): vgpr i: lanes0-15 M=i, lanes16-31 M=i+8.
// ---------------------------------------------------------------------------
__global__ void __launch_bounds__(256)
k_channel_mse(const float* __restrict__ rec, const float* __restrict__ alg,
              float* __restrict__ loss) {
  int wid  = (blockIdx.x * blockDim.x + threadIdx.x) >> 5;  // global wave id
  int lane = threadIdx.x & 31;
  int pixbase = wid * 16;                // 16 pixels per wave
  int b  = pixbase / HWSZ;
  int hw = pixbase - b * HWSZ;
  int m    = lane & 15;                  // pixel row of A handled by this lane
  int half = lane >> 4;                  // 0 -> K=0/1, 1 -> K=2/3

  const float* r0 = rec + (size_t)b * CC * HWSZ + hw + m;
  const float* g0 = alg + (size_t)b * CC * HWSZ + hw + m;

  v8f acc = {};
  v2f ones; ones[0] = 1.0f; ones[1] = 1.0f;

  for (int c0 = 0; c0 < CC; c0 += 4) {
    size_t o0 = (size_t)(c0 + 2 * half) * HWSZ;
    size_t o1 = o0 + HWSZ;
    float d0 = r0[o0] - g0[o0];
    float d1 = r0[o1] - g0[o1];
    v2f a; a[0] = d0 * d0; a[1] = d1 * d1;
    acc = __builtin_amdgcn_wmma_f32_16x16x4_f32(
        /*neg_a=*/false, a, /*neg_b=*/false, ones,
        /*c_mod=*/(short)0, acc, /*reuse_a=*/false, /*reuse_b=*/false);
  }

  // Column 0 of D: lane 0 holds pixel rows 0-7, lane 16 holds rows 8-15.
  const float s = 1.0f / (float)CC;
  if (lane == 0 || lane == 16) {
    float* out = loss + pixbase + (half ? 8 : 0);
    float4 lo = make_float4(acc[0] * s, acc[1] * s, acc[2] * s, acc[3] * s);
    float4 hi = make_float4(acc[4] * s, acc[5] * s, acc[6] * s, acc[7] * s);
    *(float4*)(out)     = lo;
    *(float4*)(out + 4) = hi;
  }
}

// ---------------------------------------------------------------------------
// Kernel 2: 3x3 disjoint block means of loss -> blk[b*1024 + hb*32 + wb]
// ---------------------------------------------------------------------------
__global__ void __launch_bounds__(256)
k_block_mean(const float* __restrict__ loss, float* __restrict__ blk) {
  int t = blockIdx.x * blockDim.x + threadIdx.x;
  if (t >= BB * NBLK) return;
  int b = t / NBLK, g = t % NBLK;
  int hb = g / NBW, wb = g % NBW;
  const float* p = loss + (size_t)b * HWSZ + (hb * RR) * WW + wb * RR;
  float s = 0.0f;
  for (int i = 0; i < RR; ++i)
    for (int j = 0; j < RR; ++j) s += p[i * WW + j];
  blk[t] = s * (1.0f / 9.0f);
}

// ---------------------------------------------------------------------------
// Kernel 3: per-batch stats. region = blk repeated 9x, so:
//   flat median (lower-middle, idx 4607) == ascending-sorted blk[511]
//   SS_flat = 9*SS_blk ; std = sqrt(9*SS/9215)  (ddof=1 over n=9216)
// Bitonic sort of 1024 values in LDS, 512 threads/batch. Fixed-order tree
// reductions -> deterministic.
// ---------------------------------------------------------------------------
__global__ void __launch_bounds__(512)
k_stats(const float* __restrict__ blk, float* __restrict__ stats) {
  __shared__ float s[NBLK];
  __shared__ float red[512];
  __shared__ float sh_mean;
  int b = blockIdx.x;
  int t = threadIdx.x;

  float a0 = blk[b * NBLK + t];
  float a1 = blk[b * NBLK + t + 512];
  s[t] = a0; s[t + 512] = a1;
  red[t] = a0 + a1;
  __syncthreads();
  for (int st = 256; st > 0; st >>= 1) {
    if (t < st) red[t] += red[t + st];
    __syncthreads();
  }
  if (t == 0) sh_mean = red[0] * (1.0f / (float)NBLK);
  __syncthreads();
  float mean = sh_mean;

  float d0 = a0 - mean, d1 = a1 - mean;
  red[t] = d0 * d0 + d1 * d1;
  __syncthreads();
  for (int st = 256; st > 0; st >>= 1) {
    if (t < st) red[t] += red[t + st];
    __syncthreads();
  }
  float SS = red[0];

  // bitonic sort s[] ascending
  for (int k = 2; k <= NBLK; k <<= 1) {
    for (int jj = k >> 1; jj > 0; jj >>= 1) {
      __syncthreads();
      int x = ((t & ~(jj - 1)) << 1) | (t & (jj - 1));
      bool up = ((x & k) == 0);
      float va = s[x], vb = s[x + jj];
      if ((va > vb) == up) { s[x] = vb; s[x + jj] = va; }
    }
  }
  __syncthreads();

  if (t == 0) {
    float med  = s[511];                       // == sorted(flat)[4607]
    float var  = 9.0f * SS / (float)(HWSZ - 1);
    float sd   = sqrtf(var);
    float safe = fmaxf(sd, EPSV);
    stats[b * 4 + 0] = med;
    stats[b * 4 + 1] = 1.0f / safe;
    stats[b * 4 + 2] = (sd > EPSV) ? 1.0f : 0.0f;
    stats[b * 4 + 3] = 0.0f;
  }
}

// ---------------------------------------------------------------------------
// Kernel 4: per-pixel focal weight + per-workgroup partial sums.
// Fallback (std<=eps, never taken for N(0,1) data) reproduces the stable
// descending-argsort rank: rank = #(greater) + #(equal with smaller flat idx).
// ---------------------------------------------------------------------------
__global__ void __launch_bounds__(256)
k_weighted(const float* __restrict__ loss, const float* __restrict__ blk,
           const float* __restrict__ stats, float* __restrict__ partial) {
  __shared__ float red[256];
  int j  = blockIdx.x * 256 + threadIdx.x;      // 288*256 == NPIX exactly
  int b  = j / HWSZ;
  int hw = j - b * HWSZ;
  int h = hw / WW, w = hw - h * WW;

  float v   = blk[b * NBLK + (h / RR) * NBW + (w / RR)];
  float med = stats[b * 4 + 0];
  float inv = stats[b * 4 + 1];
  float flg = stats[b * 4 + 2];

  float nimp;
  if (flg > 0.5f) {
    float z = (v - med) * inv;
    nimp = 1.0f / (1.0f + expf(-z));
  } else {
    const float* bp = blk + b * NBLK;
    int rank = 0;
    for (int g = 0; g < NBLK; ++g) {
      float u = bp[g];
      if (u > v) {
        rank += 9;
      } else if (u == v) {
        int hb = g / NBW, wb = g - hb * NBW;
        for (int i = 0; i < RR; ++i)
          for (int q = 0; q < RR; ++q) {
            int f = (hb * RR + i) * WW + wb * RR + q;
            rank += (f < hw) ? 1 : 0;
          }
      }
    }
    nimp = (float)(HWSZ - 1 - rank) / (float)(HWSZ - 1);
  }

  float val = loss[j] * (1.0f + BETA * nimp);

  red[threadIdx.x] = val;
  __syncthreads();
  for (int st = 128; st > 0; st >>= 1) {
    if (threadIdx.x < st) red[threadIdx.x] += red[threadIdx.x + st];
    __syncthreads();
  }
  if (threadIdx.x == 0) partial[blockIdx.x] = red[0];
}

// ---------------------------------------------------------------------------
// Kernel 5: deterministic final reduction of the 288 partials -> scalar mean.
// ---------------------------------------------------------------------------
__global__ void __launch_bounds__(512)
k_final(const float* __restrict__ partial, int n, float* __restrict__ out) {
  __shared__ float red[512];
  int t = threadIdx.x;
  red[t] = (t < n) ? partial[t] : 0.0f;
  __syncthreads();
  for (int st = 256; st > 0; st >>= 1) {
    if (t < st) red[t] += red[t + st];
    __syncthreads();
  }
  if (t == 0) out[0] = red[0] / (float)NPIX;
}

extern "C" void kernel_launch(void* const* d_in, const int* in_sizes, int n_in,
                              void* d_out, int out_size, void* d_ws, size_t ws_size,
                              hipStream_t stream) {
  const float* rec = (const float*)d_in[0];
  const float* alg = (const float*)d_in[1];
  float* ws    = (float*)d_ws;
  float* loss  = ws;                         // 73728 floats
  float* blk   = ws + NPIX;                  // 8192 floats
  float* stats = blk + BB * NBLK;            // 32 floats
  float* part  = stats + 32;                 // 288 floats
  float* out   = (float*)d_out;

  k_channel_mse<<<NPIX / 16 / 8, 256, 0, stream>>>(rec, alg, loss);  // 576 blocks
  k_block_mean <<<(BB * NBLK + 255) / 256, 256, 0, stream>>>(loss, blk);
  k_stats      <<<BB, 512, 0, stream>>>(blk, stats);
  k_weighted   <<<NPIX / 256, 256, 0, stream>>>(loss, blk, stats, part);
  k_final      <<<1, 512, 0, stream>>>(part, 288, out);
}